// SIFnode_75127567941815
// MI455X (gfx1250) — compile-verified
//
#include <hip/hip_runtime.h>

// SIF neuron scan: [T=256, HWC=196608] -> [HWC], spike-count + gamma.
// Memory-bound streaming kernel: 16-deep async global->LDS pipeline per wave
// (gfx1250 ASYNCcnt path), s_wait_asynccnt throttling, non-temporal loads,
// branchless IF update. Roofline floor ~8.7us @ 23.3 TB/s.

#define T_STEPS 256
#define DEPTH   16         // async pipeline depth (slots per thread)
#define BLOCK   256        // threads per block (8 wave32 waves)
#define EPT     2          // elements (floats) per thread -> b64 async loads

typedef __attribute__((ext_vector_type(2))) float v2f;

__global__ __launch_bounds__(BLOCK)
void sif_scan_kernel(const float* __restrict__ x, float* __restrict__ out,
                     int slice /* = H*W*C = 196608 */) {
    __shared__ v2f buf[DEPTH][BLOCK];   // 32 KB: private slot per (depth, thread)

    const int      tid         = threadIdx.x;
    const int      pos         = (blockIdx.x * BLOCK + tid) * EPT; // elem idx in slice
    const unsigned slice_bytes = (unsigned)slice * 4u;             // 786432 B stride per t

    const unsigned long long gbase = (unsigned long long)x;        // SGPR pair base
    const unsigned voff0 = (unsigned)pos * 4u;                     // per-lane byte offset

    // 32-bit LDS address of this thread's slot 0 (low 32 bits of generic ptr)
    const unsigned lds0        = (unsigned)(unsigned long long)&buf[0][tid];
    const unsigned slot_stride = (unsigned)(BLOCK * sizeof(v2f)); // 2048 B

    // ---- prologue: fill the async pipeline (loads t = 0..DEPTH-1) ----
#pragma unroll
    for (int i = 0; i < DEPTH; ++i) {
        unsigned l = lds0 + (unsigned)i * slot_stride;
        unsigned g = voff0 + (unsigned)i * slice_bytes;
        asm volatile("global_load_async_to_lds_b64 %0, %1, %2 th:TH_LOAD_NT"
                     :: "v"(l), "v"(g), "s"(gbase) : "memory");
    }

    float mem0 = 0.0f, mem1 = 0.0f;
    int   c0 = 0, c1 = 0;

    // ---- main loop: consume slot t, refill with load t+DEPTH ----
#pragma unroll 16
    for (int t = 0; t < T_STEPS - DEPTH; ++t) {
        // async loads complete in order: <=15 outstanding => load t has landed
        asm volatile("s_wait_asynccnt 15" ::: "memory");
        v2f v = buf[t & (DEPTH - 1)][tid];
        // make sure the ds_load has drained before the async engine may
        // overwrite this slot (async-LDS writes are unordered vs DS ops)
        asm volatile("s_wait_dscnt 0" ::: "memory");
        unsigned l = lds0 + (unsigned)(t & (DEPTH - 1)) * slot_stride;
        unsigned g = voff0 + (unsigned)(t + DEPTH) * slice_bytes;
        asm volatile("global_load_async_to_lds_b64 %0, %1, %2 th:TH_LOAD_NT"
                     :: "v"(l), "v"(g), "s"(gbase) : "memory");

        // integrate & fire (soft reset to exactly 0, matches mem -= ex)
        mem0 += v.x; if (mem0 > 1.0f) { ++c0; mem0 = 0.0f; }
        mem1 += v.y; if (mem1 > 1.0f) { ++c1; mem1 = 0.0f; }
    }

    // ---- epilogue: drain the last DEPTH slots ----
    asm volatile("s_wait_asynccnt 0" ::: "memory");
#pragma unroll
    for (int t = T_STEPS - DEPTH; t < T_STEPS; ++t) {
        v2f v = buf[t & (DEPTH - 1)][tid];
        mem0 += v.x; if (mem0 > 1.0f) { ++c0; mem0 = 0.0f; }
        mem1 += v.y; if (mem1 > 1.0f) { ++c1; mem1 = 0.0f; }
    }

    // y = count/256 ; out = y^(1/2.2)
    const float inv_t = 1.0f / 256.0f;
    const float gpow  = 1.0f / 2.2f;
    float y0 = (float)c0 * inv_t;
    float y1 = (float)c1 * inv_t;
    out[pos + 0] = (c0 > 0) ? __builtin_exp2f(gpow * __builtin_log2f(y0)) : 0.0f;
    out[pos + 1] = (c1 > 0) ? __builtin_exp2f(gpow * __builtin_log2f(y1)) : 0.0f;
}

extern "C" void kernel_launch(void* const* d_in, const int* in_sizes, int n_in,
                              void* d_out, int out_size, void* d_ws, size_t ws_size,
                              hipStream_t stream) {
    const float* x   = (const float*)d_in[0];   // [256, 256, 256, 3] float32
    float*       out = (float*)d_out;           // [256, 256, 3]     float32

    const int slice  = out_size;                  // 196608 elements per time slice
    const int blocks = slice / (BLOCK * EPT);     // 196608 / 512 = 384

    sif_scan_kernel<<<blocks, BLOCK, 0, stream>>>(x, out, slice);
}